// CrossLayer_20693152432611
// MI455X (gfx1250) — compile-verified
//
#include <hip/hip_runtime.h>

// CrossLayer: B=16384 rows, D=1024, L=4 layers.
// Reduces to per-row scalars s0=sum(x0), d_i=dot(w_i,x0), chained, then one
// elementwise pass. Bandwidth-bound: ~128MB traffic / 23.3 TB/s ~= 5.5us.
// Dot products done with V_WMMA_F32_16X16X4_F32 (16 rows x 4 outputs per wave).
// B-matrix columns {ones, w0, w1, w2} + a shared zero row live in LDS so the
// inner loop is pure loads + WMMA (no per-iteration cndmask masking).

typedef __attribute__((ext_vector_type(2))) float v2f;
typedef __attribute__((ext_vector_type(4))) float v4f;
typedef __attribute__((ext_vector_type(8))) float v8f;

#define NB   16384
#define DD   1024

// ---- tiny helper: Bsum[i] = sum(b[i][:]) for i=0..3, into workspace ----
__global__ __launch_bounds__(128) void bsum_kernel(const float* __restrict__ b,
                                                   float* __restrict__ ws) {
  const int lane = threadIdx.x & 31;
  const int row  = threadIdx.x >> 5;   // 0..3 (4 waves)
  float p = 0.0f;
  for (int j = lane; j < DD; j += 32) p += b[row * DD + j];
#pragma unroll
  for (int s = 16; s; s >>= 1) p += __shfl_down(p, s, 32);
  if (lane == 0) ws[row] = p;
}

// ---- main kernel: one wave handles 16 rows ----
__global__ __launch_bounds__(256) void cross_main(const float* __restrict__ x,
                                                  const float* __restrict__ w,
                                                  const float* __restrict__ b,
                                                  const float* __restrict__ bsums,
                                                  float* __restrict__ out) {
  __shared__ float Bmat[5 * DD];   // rows: ones, w0, w1, w2, zeros  (20 KB)
  __shared__ float S[8 * 256];     // per-wave C dump: [wave][n*16+m]  (8 KB)

  const int tid = threadIdx.x;

  // Stage B-matrix rows {ones, w0, w1, w2, zeros} into LDS (coalesced).
  for (int idx = tid; idx < 5 * DD; idx += 256) {
    const int r = idx >> 10;
    const int c = idx & (DD - 1);
    float v;
    if (r == 0)      v = 1.0f;
    else if (r <= 3) v = w[(r - 1) * DD + c];
    else             v = 0.0f;
    Bmat[idx] = v;
  }
  __syncthreads();

  const int lane = tid & 31;
  const int wv   = tid >> 5;           // wave in block, 0..7
  const int q    = lane & 15;          // row index (A) / col index (B)
  const int half = lane >> 4;          // k-offset selector: 0 -> k,k+1 ; 1 -> k+2,k+3
  const int r0   = (blockIdx.x * 8 + wv) * 16;

  // A fragment source: lane covers row r0+q, columns k + 2*half, k + 2*half + 1.
  const float* xlane = x + (size_t)(r0 + q) * DD + 2 * half;
  // B fragment source: column q of Bmat; lanes q>=4 all read the shared zero
  // row (same-address LDS broadcast, no bank conflict, no cndmask needed).
  const int    qb    = (q < 4) ? q : 4;
  const float* blane = &Bmat[qb * DD + 2 * half];

  // 4 interleaved accumulators to break the WMMA C->C dependency chain.
  v8f acc[4];
#pragma unroll
  for (int u = 0; u < 4; ++u) {
    v8f z = {0.f, 0.f, 0.f, 0.f, 0.f, 0.f, 0.f, 0.f};
    acc[u] = z;
  }

  for (int k0 = 0; k0 < DD; k0 += 16) {
#pragma unroll
    for (int u = 0; u < 4; ++u) {
      const int k = k0 + 4 * u;
      const v2f a  = *(const v2f*)(xlane + k);
      const v2f bb = *(const v2f*)(blane + k);
      acc[u] = __builtin_amdgcn_wmma_f32_16x16x4_f32(
          /*neg_a=*/false, a, /*neg_b=*/false, bb,
          /*c_mod=*/(short)0, acc[u], /*reuse_a=*/false, /*reuse_b=*/false);
    }
  }
  const v8f c = (acc[0] + acc[1]) + (acc[2] + acc[3]);

  // Dump C[m][n] to LDS: lane (q,half) holds rows m = half*8 + j, col n = q.
  float* Sw = &S[wv * 256];
  {
    const int base = q * 16 + half * 8;
#pragma unroll
    for (int j = 0; j < 8; ++j) Sw[base + j] = c[j];
  }
  __syncthreads();

  const float Bs0 = bsums[0];
  const float Bs1 = bsums[1];
  const float Bs2 = bsums[2];
  const float* w3 = w + 3 * DD;
  const float* b3 = b + 3 * DD;

  // Per-row scalar chain + streaming elementwise output.
  for (int m = 0; m < 16; ++m) {
    const float s0 = Sw[0 * 16 + m];
    const float d0 = Sw[1 * 16 + m];
    const float d1 = Sw[2 * 16 + m];
    const float d2 = Sw[3 * 16 + m];
    const float s1 = fmaf(s0, d0, Bs0 + s0);
    const float s2 = fmaf(s1, d1, Bs1 + s1);
    const float s3 = fmaf(s2, d2, Bs2 + s2);

    const float* xr   = x   + (size_t)(r0 + m) * DD;
    float*       orow = out + (size_t)(r0 + m) * DD;
#pragma unroll
    for (int t = 0; t < 8; ++t) {
      const int j = t * 128 + lane * 4;
      const v4f xv = *(const v4f*)(xr + j);
      const v4f wvv = *(const v4f*)(w3 + j);
      const v4f bvv = *(const v4f*)(b3 + j);
      v4f ov;
#pragma unroll
      for (int e = 0; e < 4; ++e) ov[e] = fmaf(s3 * wvv[e], xv[e], bvv[e] + xv[e]);
      *(v4f*)(orow + j) = ov;
    }
  }
}

extern "C" void kernel_launch(void* const* d_in, const int* in_sizes, int n_in,
                              void* d_out, int out_size, void* d_ws, size_t ws_size,
                              hipStream_t stream) {
  const float* x = (const float*)d_in[0];   // (16384, 1, 1024) f32
  const float* w = (const float*)d_in[1];   // (4, 1024) f32
  const float* b = (const float*)d_in[2];   // (4, 1024) f32
  float* out = (float*)d_out;               // (16384, 1024) f32
  float* ws  = (float*)d_ws;                // >= 16 bytes used

  bsum_kernel<<<1, 128, 0, stream>>>(b, ws);
  // 16 rows/wave, 8 waves/block -> 128 rows/block -> 128 blocks.
  cross_main<<<NB / 128, 256, 0, stream>>>(x, w, b, ws, out);
}